// OnlineMemoryAttention_15109694947379
// MI455X (gfx1250) — compile-verified
//
#include <hip/hip_runtime.h>
#include <hip/hip_bf16.h>

typedef __attribute__((ext_vector_type(2))) float    v2f;
typedef __attribute__((ext_vector_type(8))) float    v8f;
typedef __attribute__((ext_vector_type(4))) unsigned v4u;
typedef __attribute__((ext_vector_type(8))) int      v8i;
typedef __attribute__((ext_vector_type(4))) int      v4i;

#define T_SEQ 2048
#define D_MOD 1024
#define NH    16
#define DH    64
#define LN_DECAY (-0.010050335853501441f)   // ln(0.99)

// ---- Tensor Data Mover availability ---------------------------------------
#if defined(__has_builtin)
#  if __has_builtin(__builtin_amdgcn_tensor_load_to_lds)
#    define USE_TDM 1
#  endif
#endif
#ifndef USE_TDM
#  define USE_TDM 0
#endif

__device__ __forceinline__ void wait_tensorcnt0() {
#if defined(__has_builtin) && __has_builtin(__builtin_amdgcn_s_wait_tensorcnt)
    __builtin_amdgcn_s_wait_tensorcnt(0);
#else
    asm volatile("s_wait_tensorcnt 0x0" ::: "memory");
#endif
}

#if USE_TDM
typedef __attribute__((address_space(3))) float lds_float;

// 2-D TDM descriptor per CDNA5 ISA (D# groups 0/1; groups
// 2/3 zero => 2-D tensor). data_size=4B (code 2), no padding, no multicast.
// Toolchain uses the 6-arg builtin: (v4u, v8i, v4i, v4i, v8i, i32 cpol).
__device__ __forceinline__ void tdm_load_tile_2d(
    unsigned lds_byte, const float* gsrc,
    unsigned tensor_w, unsigned tensor_h, unsigned stride_w,
    unsigned tile_w, unsigned tile_h)
{
    unsigned long long ga = (unsigned long long)(size_t)gsrc;
    v4u g0;
    g0.x = 1u;                                            // count=1 (valid user D#)
    g0.y = lds_byte;                                      // lds_addr
    g0.z = (unsigned)ga;                                  // global_addr[95:64]
    g0.w = (unsigned)((ga >> 32) & 0x01FFFFFFull)         // global_addr[120:96]
         | (2u << 30);                                    // type=2 ("image")
    v8i g1;
    g1[0] = (int)(2u << 16);                              // data_size = 4 bytes
    g1[1] = (int)((tensor_w & 0xFFFFu) << 16);            // tensor_dim0[15:0]
    g1[2] = (int)((tensor_w >> 16) | ((tensor_h & 0xFFFFu) << 16)); // dim0 hi | dim1 lo
    g1[3] = (int)((tensor_h >> 16) | (tile_w << 16));     // dim1 hi | tile_dim0
    g1[4] = (int)tile_h;                                  // tile_dim1 (tile_dim2=0)
    g1[5] = (int)stride_w;                                // tensor_dim0_stride lo
    g1[6] = 0;                                            // stride0 hi | stride1 lo
    g1[7] = 0;                                            // stride1 hi
    v4i gz4 = {0, 0, 0, 0};
    v8i gz8 = {0, 0, 0, 0, 0, 0, 0, 0};
    __builtin_amdgcn_tensor_load_to_lds(g0, g1, gz4, gz4, gz8, 0);
}
#endif

// ---------------------------------------------------------------------------
// Tiled GEMM: C[M,N] = A[M,K] @ B[K,N] (+bias), fp32 via WMMA 16x16x4.
// Block = 128 threads (4 waves); block tile 64 x (NT*16); wave w owns rows
// [16w,16w+16) x all NT column tiles. K chunked by 16; B chunk staged in LDS
// (TDM double-buffered DMA when available). All dims are exact multiples here,
// so the inner loop is completely guard-free.
// A-frag: lane holds A[row0+lid][k+2*half+{0,1}]
// B-frag: lane holds B[k+2*half+{0,1}][col0+lid]
// C tile: lane(l), vgpr(r) -> C[row0 + r + 8*half][col0 + lid]
// ---------------------------------------------------------------------------
template <int NT>
__global__ __launch_bounds__(128) void gemm_wmma_f32(
    const float* __restrict__ A, const float* __restrict__ B,
    const float* __restrict__ bias, float* __restrict__ C,
    int N, int K)
{
    constexpr int BW = NT * 16;                 // B tile width
    __shared__ float bsh[2][16 * BW];           // double-buffered K-chunk of B
    const int wave = threadIdx.x >> 5;
    const int lane = threadIdx.x & 31;
    const int half = lane >> 4;
    const int lid  = lane & 15;
    const int row0 = blockIdx.x * 64 + wave * 16;
    const int col0 = blockIdx.y * BW;
    const int nchunks = K / 16;

#if USE_TDM
    const unsigned lds0 = (unsigned)(size_t)(lds_float*)&bsh[0][0];
    const unsigned lbsz = (unsigned)(16 * BW * sizeof(float));
    if (wave == 0) {
        tdm_load_tile_2d(lds0, B + col0, (unsigned)N, (unsigned)K,
                         (unsigned)N, BW, 16);
        wait_tensorcnt0();
    }
    __syncthreads();
#endif

    v8f acc[NT] = {};
    for (int c = 0; c < nchunks; ++c) {
        const int k0 = c * 16;
#if USE_TDM
        if (wave == 0 && c + 1 < nchunks)       // prefetch next chunk via TDM
            tdm_load_tile_2d(lds0 + ((unsigned)(c + 1) & 1u) * lbsz,
                             B + (size_t)(k0 + 16) * N + col0,
                             (unsigned)N, (unsigned)K, (unsigned)N, BW, 16);
        const float* bb = bsh[c & 1];
#else
        __syncthreads();
        for (int i = threadIdx.x; i < 16 * BW; i += 128)
            bsh[0][i] = B[(size_t)(k0 + i / BW) * N + col0 + (i % BW)];
        __syncthreads();
        const float* bb = bsh[0];
#endif
        // A frags for this chunk (4 K-steps of 4)
        v2f a[4];
#pragma unroll
        for (int ks = 0; ks < 4; ++ks) {
            const float* ap = A + (size_t)(row0 + lid) * K + k0 + ks * 4 + 2 * half;
            a[ks].x = ap[0]; a[ks].y = ap[1];
        }
#pragma unroll
        for (int t = 0; t < NT; ++t) {
#pragma unroll
            for (int ks = 0; ks < 4; ++ks) {
                v2f b;
                b.x = bb[(ks * 4 + 2 * half)     * BW + t * 16 + lid];
                b.y = bb[(ks * 4 + 2 * half + 1) * BW + t * 16 + lid];
                acc[t] = __builtin_amdgcn_wmma_f32_16x16x4_f32(
                    false, a[ks], false, b, (short)0, acc[t], false, false);
            }
        }
#if USE_TDM
        if (wave == 0 && c + 1 < nchunks) wait_tensorcnt0();
        __syncthreads();
#endif
    }
#pragma unroll
    for (int t = 0; t < NT; ++t) {
        const int c = col0 + t * 16 + lid;
        const float bv = bias ? bias[c] : 0.f;
#pragma unroll
        for (int r = 0; r < 8; ++r)
            C[(size_t)(row0 + r + 8 * half) * N + c] = acc[t][r] + bv;
    }
}

// ---------------------------------------------------------------------------
// Flash-style causal attention, one wave per (head, 16-row query tile).
// qkv layout: [T, 3, H, DH]   (row stride 3*H*DH);  out layout: [T, H*DH]
// ---------------------------------------------------------------------------
__global__ __launch_bounds__(32) void attn_wmma(
    const float* __restrict__ qkv, float* __restrict__ out)
{
    __shared__ float pbuf[16 * 17];
    const int lane = threadIdx.x & 31;
    const int half = lane >> 4;
    const int lid  = lane & 15;
    const int t0   = blockIdx.x * 16;
    const int h    = blockIdx.y;
    const size_t tstr = (size_t)3 * NH * DH;
    const float scale = 0.125f;                   // dh^-0.5
    const float* Q  = qkv + (size_t)h * DH;
    const float* Kp = qkv + (size_t)NH * DH + (size_t)h * DH;
    const float* Vp = qkv + 2 * (size_t)NH * DH + (size_t)h * DH;

    v2f qa[16];                                   // Q tile as 16 A-frags (K=DH)
#pragma unroll
    for (int kk = 0; kk < 16; ++kk) {
        const float* qp = Q + (size_t)(t0 + lid) * tstr + kk * 4 + 2 * half;
        qa[kk].x = qp[0]; qa[kk].y = qp[1];
    }

    v8f o[4] = {};
    float m[8], l[8];
#pragma unroll
    for (int r = 0; r < 8; ++r) { m[r] = -3.0e38f; l[r] = 0.f; }

    for (int s0 = 0; s0 <= t0; s0 += 16) {
        v8f s = {};
#pragma unroll
        for (int kk = 0; kk < 16; ++kk) {         // S = Q @ K^T
            const float* kp = Kp + (size_t)(s0 + lid) * tstr + kk * 4 + 2 * half;
            v2f b; b.x = kp[0]; b.y = kp[1];
            s = __builtin_amdgcn_wmma_f32_16x16x4_f32(
                false, qa[kk], false, b, (short)0, s, false, false);
        }
        const bool diag = (s0 == t0);
#pragma unroll
        for (int r = 0; r < 8; ++r) {
            float v = s[r] * scale;
            if (diag && (lid > r + 8 * half)) v = -3.0e38f;   // strict future
            float rm = v;
#pragma unroll
            for (int off = 1; off < 16; off <<= 1)
                rm = fmaxf(rm, __shfl_xor(rm, off, 32));
            const float mn   = fmaxf(m[r], rm);
            const float corr = __expf(m[r] - mn);
            const float pv   = __expf(v - mn);
            float rs = pv;
#pragma unroll
            for (int off = 1; off < 16; off <<= 1)
                rs += __shfl_xor(rs, off, 32);
            l[r] = l[r] * corr + rs;
            m[r] = mn;
#pragma unroll
            for (int t = 0; t < 4; ++t) o[t][r] *= corr;
            pbuf[(r + 8 * half) * 17 + lid] = pv;   // stage P (C-layout -> LDS)
        }
        __syncthreads();
        v2f pa[4];                                  // P in A-frag layout
#pragma unroll
        for (int kk = 0; kk < 4; ++kk) {
            pa[kk].x = pbuf[lid * 17 + kk * 4 + 2 * half];
            pa[kk].y = pbuf[lid * 17 + kk * 4 + 2 * half + 1];
        }
        __syncthreads();
#pragma unroll
        for (int t = 0; t < 4; ++t) {               // O += P @ V
#pragma unroll
            for (int kk = 0; kk < 4; ++kk) {
                const float* vp = Vp + (size_t)(s0 + kk * 4 + 2 * half) * tstr + t * 16 + lid;
                v2f b; b.x = vp[0]; b.y = vp[tstr];
                o[t] = __builtin_amdgcn_wmma_f32_16x16x4_f32(
                    false, pa[kk], false, b, (short)0, o[t], false, false);
            }
        }
    }
#pragma unroll
    for (int r = 0; r < 8; ++r) l[r] = 1.f / l[r];
#pragma unroll
    for (int t = 0; t < 4; ++t)
#pragma unroll
        for (int r = 0; r < 8; ++r)
            out[(size_t)(t0 + r + 8 * half) * (NH * DH) + h * DH + t * 16 + lid] =
                o[t][r] * l[r];
}

// ---------------------------------------------------------------------------
// Plucker lines: exterior product + normalize; u = J6 @ write_line is just
// (e5,-e4,e3,e2,-e1,e0). Stored [H][T][8] zero-padded so incidence WMMA K=8.
// ---------------------------------------------------------------------------
__device__ inline void exterior6(const float a[4], const float b[4], float e[6]) {
    e[0] = a[0]*b[1] - a[1]*b[0];
    e[1] = a[0]*b[2] - a[2]*b[0];
    e[2] = a[0]*b[3] - a[3]*b[0];
    e[3] = a[1]*b[2] - a[2]*b[1];
    e[4] = a[1]*b[3] - a[3]*b[1];
    e[5] = a[2]*b[3] - a[3]*b[2];
    float n = 0.f;
#pragma unroll
    for (int i = 0; i < 6; ++i) n += e[i]*e[i];
    const float inv = 1.f / fmaxf(sqrtf(n), 1e-12f);
#pragma unroll
    for (int i = 0; i < 6; ++i) e[i] *= inv;
}

__global__ void lines_kernel(
    const float* __restrict__ pw1, const float* __restrict__ pw2,
    const float* __restrict__ pr1, const float* __restrict__ pr2,
    float* __restrict__ u, float* __restrict__ rl)
{
    const int i = blockIdx.x * blockDim.x + threadIdx.x;
    if (i >= T_SEQ * NH) return;
    const int t = i / NH, h = i % NH;
    float a[4], b[4], e[6];
#pragma unroll
    for (int d = 0; d < 4; ++d) {
        a[d] = (t == 0) ? 0.f : pw1[(size_t)(t - 1) * (NH * 4) + h * 4 + d];
        b[d] = pw2[(size_t)t * (NH * 4) + h * 4 + d];
    }
    exterior6(a, b, e);
    const size_t base = ((size_t)h * T_SEQ + t) * 8;
    u[base + 0] =  e[5]; u[base + 1] = -e[4]; u[base + 2] =  e[3];
    u[base + 3] =  e[2]; u[base + 4] = -e[1]; u[base + 5] =  e[0];
    u[base + 6] = 0.f;   u[base + 7] = 0.f;
#pragma unroll
    for (int d = 0; d < 4; ++d) {
        a[d] = pr1[(size_t)t * (NH * 4) + h * 4 + d];
        b[d] = pr2[(size_t)t * (NH * 4) + h * 4 + d];
    }
    exterior6(a, b, e);
#pragma unroll
    for (int d = 0; d < 6; ++d) rl[base + d] = e[d];
    rl[base + 6] = 0.f; rl[base + 7] = 0.f;
}

// ---------------------------------------------------------------------------
// mem_score[h][t] = sum_{s<t} decay^(t-s) * (rl_t . u_s)^2
// One wave per (head, 16-row t-tile); incidence 16x16 tile = 2 WMMAs (K=8).
// ---------------------------------------------------------------------------
__global__ __launch_bounds__(32) void memscore_wmma(
    const float* __restrict__ rl, const float* __restrict__ u,
    float* __restrict__ ms)
{
    const int lane = threadIdx.x & 31;
    const int half = lane >> 4;
    const int lid  = lane & 15;
    const int t0   = blockIdx.x * 16;
    const int h    = blockIdx.y;
    const float* R = rl + (size_t)h * T_SEQ * 8;
    const float* U = u  + (size_t)h * T_SEQ * 8;

    v2f ra[2];
#pragma unroll
    for (int kk = 0; kk < 2; ++kk) {
        const float* rp = R + (size_t)(t0 + lid) * 8 + kk * 4 + 2 * half;
        ra[kk].x = rp[0]; ra[kk].y = rp[1];
    }
    float acc[8] = {};
    for (int s0 = 0; s0 <= t0; s0 += 16) {
        v8f s = {};
#pragma unroll
        for (int kk = 0; kk < 2; ++kk) {
            const float* up = U + (size_t)(s0 + lid) * 8 + kk * 4 + 2 * half;
            v2f b; b.x = up[0]; b.y = up[1];
            s = __builtin_amdgcn_wmma_f32_16x16x4_f32(
                false, ra[kk], false, b, (short)0, s, false, false);
        }
#pragma unroll
        for (int r = 0; r < 8; ++r) {
            const int dt = (t0 + r + 8 * half) - (s0 + lid);
            if (dt > 0)
                acc[r] += s[r] * s[r] * __expf((float)dt * LN_DECAY);
        }
    }
#pragma unroll
    for (int r = 0; r < 8; ++r) {
#pragma unroll
        for (int off = 1; off < 16; off <<= 1)
            acc[r] += __shfl_xor(acc[r], off, 32);
    }
    if (lid == 0)
#pragma unroll
        for (int r = 0; r < 8; ++r)
            ms[(size_t)h * T_SEQ + t0 + r + 8 * half] = acc[r];
}

// ---------------------------------------------------------------------------
// y[t] = seq_out[t] + mean_h( sigmoid(ms[h][t]*scale[h]) * sigmoid(memg[t][h]) )
//        * mem_val[t]
// ---------------------------------------------------------------------------
__global__ void combine_kernel(
    const float* __restrict__ seq, const float* __restrict__ memval,
    const float* __restrict__ memg, const float* __restrict__ ms,
    const float* __restrict__ mem_scale, float* __restrict__ y)
{
    const int t = blockIdx.x;
    __shared__ float gsh;
    if (threadIdx.x == 0) {
        float acc = 0.f;
        for (int h = 0; h < NH; ++h) {
            const float sm = 1.f / (1.f + __expf(-ms[(size_t)h * T_SEQ + t] * mem_scale[h]));
            const float gg = 1.f / (1.f + __expf(-memg[(size_t)t * NH + h]));
            acc += sm * gg;
        }
        gsh = acc * (1.0f / NH);
    }
    __syncthreads();
    const float g = gsh;
    for (int d = threadIdx.x; d < D_MOD; d += blockDim.x)
        y[(size_t)t * D_MOD + d] =
            seq[(size_t)t * D_MOD + d] + g * memval[(size_t)t * D_MOD + d];
}

// ---------------------------------------------------------------------------
extern "C" void kernel_launch(void* const* d_in, const int* in_sizes, int n_in,
                              void* d_out, int out_size, void* d_ws, size_t ws_size,
                              hipStream_t stream)
{
    const float* x        = (const float*)d_in[0];
    const float* qkv_w    = (const float*)d_in[1];
    const float* qkv_b    = (const float*)d_in[2];
    const float* w1w      = (const float*)d_in[3];
    const float* w2w      = (const float*)d_in[4];
    const float* r1w      = (const float*)d_in[5];
    const float* r2w      = (const float*)d_in[6];
    const float* memv_w   = (const float*)d_in[7];
    const float* memv_b   = (const float*)d_in[8];
    const float* memg_w   = (const float*)d_in[9];
    const float* memg_b   = (const float*)d_in[10];
    const float* memscale = (const float*)d_in[11];
    const float* out_w    = (const float*)d_in[12];
    const float* out_b    = (const float*)d_in[13];
    float* out = (float*)d_out;

    float* ws = (float*)d_ws;
    float* qkv    = ws;                                   // T * 3D
    float* pw1    = qkv    + (size_t)T_SEQ * 3 * D_MOD;   // T*64 each
    float* pw2    = pw1    + (size_t)T_SEQ * 64;
    float* pr1    = pw2    + (size_t)T_SEQ * 64;
    float* pr2    = pr1    + (size_t)T_SEQ * 64;
    float* memval = pr2    + (size_t)T_SEQ * 64;          // T*D
    float* memg   = memval + (size_t)T_SEQ * D_MOD;       // T*H
    float* ubuf   = memg   + (size_t)T_SEQ * NH;          // H*T*8
    float* rlbuf  = ubuf   + (size_t)NH * T_SEQ * 8;      // H*T*8
    float* msbuf  = rlbuf  + (size_t)NH * T_SEQ * 8;      // H*T
    float* seq    = msbuf  + (size_t)NH * T_SEQ;          // T*D
    float* ybuf   = seq    + (size_t)T_SEQ * D_MOD;       // T*D

    const dim3 blk128(128), blk32(32), blk256(256);

    // 1) qkv = x @ qkv_w + qkv_b
    gemm_wmma_f32<4><<<dim3(T_SEQ / 64, 3 * D_MOD / 64), blk128, 0, stream>>>(
        x, qkv_w, qkv_b, qkv, 3 * D_MOD, D_MOD);

    // 2) point projections (N = 64)
    gemm_wmma_f32<4><<<dim3(T_SEQ / 64, 1), blk128, 0, stream>>>(x, w1w, nullptr, pw1, 64, D_MOD);
    gemm_wmma_f32<4><<<dim3(T_SEQ / 64, 1), blk128, 0, stream>>>(x, w2w, nullptr, pw2, 64, D_MOD);
    gemm_wmma_f32<4><<<dim3(T_SEQ / 64, 1), blk128, 0, stream>>>(x, r1w, nullptr, pr1, 64, D_MOD);
    gemm_wmma_f32<4><<<dim3(T_SEQ / 64, 1), blk128, 0, stream>>>(x, r2w, nullptr, pr2, 64, D_MOD);

    // 3) mem value / gate projections
    gemm_wmma_f32<4><<<dim3(T_SEQ / 64, D_MOD / 64), blk128, 0, stream>>>(
        x, memv_w, memv_b, memval, D_MOD, D_MOD);
    gemm_wmma_f32<1><<<dim3(T_SEQ / 64, 1), blk128, 0, stream>>>(
        x, memg_w, memg_b, memg, NH, D_MOD);

    // 4) Plucker lines (+ J6 application) into padded [H][T][8]
    lines_kernel<<<dim3((T_SEQ * NH + 255) / 256), blk256, 0, stream>>>(
        pw1, pw2, pr1, pr2, ubuf, rlbuf);

    // 5) causal attention -> seq_out
    attn_wmma<<<dim3(T_SEQ / 16, NH), blk32, 0, stream>>>(qkv, seq);

    // 6) decay-weighted squared incidence row sums -> mem_score
    memscore_wmma<<<dim3(T_SEQ / 16, NH), blk32, 0, stream>>>(rlbuf, ubuf, msbuf);

    // 7) gate + combine -> y
    combine_kernel<<<dim3(T_SEQ), blk256, 0, stream>>>(
        seq, memval, memg, msbuf, memscale, ybuf);

    // 8) out = y @ out_w + out_b
    gemm_wmma_f32<4><<<dim3(T_SEQ / 64, D_MOD / 64), blk128, 0, stream>>>(
        ybuf, out_w, out_b, out, D_MOD, D_MOD);
}